// SoftSplat_51908974739591
// MI455X (gfx1250) — compile-verified
//
#include <hip/hip_runtime.h>
#include <math.h>

// Native 16-byte vector type (accepted by __builtin_nontemporal_* builtins).
typedef float v4f __attribute__((ext_vector_type(4)));

// Problem constants from the reference (fixed shapes).
constexpr int N  = 2;
constexpr int C  = 64;
constexpr int H  = 540;
constexpr int W  = 960;
constexpr int HW = H * W;                 // 518,400
constexpr int PIXELS = N * HW;            // 1,036,800
constexpr int CHUNK  = 16;                // channels per thread
constexpr float EPS  = 1e-07f;

// ---------------------------------------------------------------------------
// Kernel 1: zero the accumulator (d_out) and the weight plane (d_ws).
// 16-byte stores; W divisible by 4 so everything is aligned.
// ---------------------------------------------------------------------------
__global__ __launch_bounds__(256) void zero_kernel(v4f* __restrict__ acc4,
                                                   v4f* __restrict__ w4,
                                                   int n_acc4, int n_w4) {
    int i = blockIdx.x * blockDim.x + threadIdx.x;
    v4f z = {0.f, 0.f, 0.f, 0.f};
    if (i < n_acc4) acc4[i] = z;
    if (i < n_w4)   w4[i]   = z;
}

// ---------------------------------------------------------------------------
// Kernel 2: softmax splat with native global_atomic_add_f32.
// grid.x covers pixels (coalesced), grid.y covers channel chunks.
// ---------------------------------------------------------------------------
__global__ __launch_bounds__(256) void splat_kernel(const float* __restrict__ tenIn,
                                                    const float* __restrict__ tenFlow,
                                                    const float* __restrict__ tenMetric,
                                                    float* __restrict__ acc,
                                                    float* __restrict__ wacc) {
    const int p = blockIdx.x * blockDim.x + threadIdx.x;
    if (p >= PIXELS) return;

    const int n = p / HW;
    const int s = p - n * HW;          // spatial index y*W + x
    const int y = s / W;
    const int x = s - y * W;

    const float fx = (float)x + tenFlow[(n * 2 + 0) * HW + s];
    const float fy = (float)y + tenFlow[(n * 2 + 1) * HW + s];
    if (!(isfinite(fx) && isfinite(fy))) return;   // masked weights -> zero contribution

    const float m = expf(tenMetric[n * HW + s]);

    const float nwX = floorf(fx);
    const float nwY = floorf(fy);
    const float ax  = fx - nwX;        // in [0,1)
    const float ay  = fy - nwY;
    const float wNW = (1.f - ax) * (1.f - ay);
    const float wNE = ax * (1.f - ay);
    const float wSW = (1.f - ax) * ay;
    const float wSE = ax * ay;

    // Per-axis validity in float (avoids int-cast UB on wild values).
    const bool vx0 = (nwX       >= 0.f) && (nwX       < (float)W);
    const bool vx1 = (nwX + 1.f >= 0.f) && (nwX + 1.f < (float)W);
    const bool vy0 = (nwY       >= 0.f) && (nwY       < (float)H);
    const bool vy1 = (nwY + 1.f >= 0.f) && (nwY + 1.f < (float)H);
    const bool v00 = vx0 && vy0, v10 = vx1 && vy0;
    const bool v01 = vx0 && vy1, v11 = vx1 && vy1;
    if (!(v00 | v10 | v01 | v11)) return;

    // Safe int conversion after clamping to a benign range.
    const int ix = (int)fmaxf(fminf(nwX, (float)W), -2.f);
    const int iy = (int)fmaxf(fminf(nwY, (float)H), -2.f);

    const long long o00 = (long long)iy * W + ix;  // may be negative; only used if valid
    const long long o10 = o00 + 1;
    const long long o01 = o00 + W;
    const long long o11 = o00 + W + 1;

    // Weight (normalization) channel: exactly one chunk does it.
    if (blockIdx.y == 0) {
        float* wp = wacc + (long long)n * HW;
        if (v00) unsafeAtomicAdd(wp + o00, wNW * m);
        if (v10) unsafeAtomicAdd(wp + o10, wNE * m);
        if (v01) unsafeAtomicAdd(wp + o01, wSW * m);
        if (v11) unsafeAtomicAdd(wp + o11, wSE * m);
    }

    const int cbase = blockIdx.y * CHUNK;
    const float* ip = tenIn + ((long long)(n * C + cbase)) * HW + s;
    float*       ap = acc   + ((long long)(n * C + cbase)) * HW;

#pragma unroll
    for (int c = 0; c < CHUNK; ++c) {
        // gfx1250: prefetch the next channel (stride H*W) -> global_prefetch_b8
        __builtin_prefetch(ip + HW, 0, 1);
        // tenIn is read exactly once: non-temporal load keeps L2 for the
        // atomic accumulator working set.
        const float v = __builtin_nontemporal_load(ip) * m;
        if (v00) unsafeAtomicAdd(ap + o00, v * wNW);
        if (v10) unsafeAtomicAdd(ap + o10, v * wNE);
        if (v01) unsafeAtomicAdd(ap + o01, v * wSW);
        if (v11) unsafeAtomicAdd(ap + o11, v * wSE);
        ip += HW;
        ap += HW;
    }
}

// ---------------------------------------------------------------------------
// Kernel 3: in-place normalization  out = acc / (w + EPS), 4-wide vectorized,
// non-temporal stores (output is write-once, never re-read).
// ---------------------------------------------------------------------------
__global__ __launch_bounds__(256) void normalize_kernel(float* __restrict__ acc,
                                                        const float* __restrict__ wacc) {
    const int total4 = (N * C * HW) / 4;
    const int i = blockIdx.x * blockDim.x + threadIdx.x;
    if (i >= total4) return;

    const long long e  = (long long)i * 4;
    const int nc = (int)(e / HW);
    const int s  = (int)(e - (long long)nc * HW);
    const int n  = nc / C;

    v4f a = *(const v4f*)(acc + e);
    const v4f w = *(const v4f*)(wacc + (long long)n * HW + s);
    a.x = a.x / (w.x + EPS);
    a.y = a.y / (w.y + EPS);
    a.z = a.z / (w.z + EPS);
    a.w = a.w / (w.w + EPS);
    __builtin_nontemporal_store(a, (v4f*)(acc + e));
}

// ---------------------------------------------------------------------------
extern "C" void kernel_launch(void* const* d_in, const int* in_sizes, int n_in,
                              void* d_out, int out_size, void* d_ws, size_t ws_size,
                              hipStream_t stream) {
    const float* tenIn     = (const float*)d_in[0];   // [N,C,H,W]
    const float* tenFlow   = (const float*)d_in[1];   // [N,2,H,W]
    const float* tenMetric = (const float*)d_in[2];   // [N,1,H,W]
    float* acc  = (float*)d_out;                      // accumulate directly into output
    float* wacc = (float*)d_ws;                       // [N,H,W] splatted weights

    (void)in_sizes; (void)n_in; (void)out_size; (void)ws_size;

    const int n_acc4 = (N * C * HW) / 4;              // 16,588,800
    const int n_w4   = (N * HW) / 4;                  // 259,200

    {
        dim3 block(256);
        dim3 grid((n_acc4 + 255) / 256);
        zero_kernel<<<grid, block, 0, stream>>>((v4f*)acc, (v4f*)wacc, n_acc4, n_w4);
    }
    {
        dim3 block(256);
        dim3 grid((PIXELS + 255) / 256, C / CHUNK);   // (4050, 4)
        splat_kernel<<<grid, block, 0, stream>>>(tenIn, tenFlow, tenMetric, acc, wacc);
    }
    {
        dim3 block(256);
        dim3 grid((n_acc4 + 255) / 256);
        normalize_kernel<<<grid, block, 0, stream>>>(acc, wacc);
    }
}